// CrystalGNN_88055419503291
// MI455X (gfx1250) — compile-verified
//
#include <hip/hip_runtime.h>

#define NN 100000
#define NE 3200000
#define NG 1024
#define H  64
#define EPSV 1e-5f

typedef __attribute__((ext_vector_type(2))) float v2f;
typedef __attribute__((ext_vector_type(8))) float v8f;

// ---------------- degree / dinv ----------------
__global__ void k_init_deg(float* deg) {
    int i = blockIdx.x * blockDim.x + threadIdx.x;
    if (i < NN) deg[i] = 1.0f;              // self-loop contributes 1
}
__global__ void k_deg_accum(const int* __restrict__ ei, float* deg) {
    int e = blockIdx.x * blockDim.x + threadIdx.x;
    if (e < NE) atomicAdd(&deg[ei[NE + e]], 1.0f);
}
__global__ void k_dinv(float* deg) {
    int i = blockIdx.x * blockDim.x + threadIdx.x;
    if (i < NN) deg[i] = rsqrtf(deg[i]);    // deg >= 1 always
}

// ---------------- layer-1 GEMM: [N,3] @ [3,64] ----------------
__global__ void k_gemm3(const float* __restrict__ x, const float* __restrict__ W,
                        float* __restrict__ h) {
    int idx = blockIdx.x * blockDim.x + threadIdx.x;
    if (idx >= NN * H) return;
    int i = idx >> 6, c = idx & 63;
    h[idx] = x[i*3+0]*W[c] + x[i*3+1]*W[64+c] + x[i*3+2]*W[128+c];
}

// ---------------- WMMA GEMM: [N,64] @ [64,64], f32 exact ----------------
// One wave (32 lanes) per 16-row tile. 16 K-chunks of 4 x 4 N-tiles = 64 WMMAs.
__global__ __launch_bounds__(32) void k_gemm64_wmma(const float* __restrict__ X,
                                                    const float* __restrict__ W,
                                                    float* __restrict__ Hout) {
    const int tile = blockIdx.x;          // 6250 tiles (100000/16 exactly)
    const int lane = threadIdx.x;
    const int row  = lane & 15;
    const int kb   = (lane >> 4) * 2;     // K sub-offset per half-wave (0 or 2)
    const float* xrow = X + (tile * 16 + row) * H;

    v8f acc0 = {}, acc1 = {}, acc2 = {}, acc3 = {};
    #pragma unroll
    for (int kk = 0; kk < 16; ++kk) {
        const int k0 = kk * 4 + kb;
        v2f a; a.x = xrow[k0]; a.y = xrow[k0 + 1];
        const float* w0 = W + k0 * H;         // row K=k0
        const float* w1 = W + (k0 + 1) * H;   // row K=k0+1
        v2f b0, b1, b2, b3;
        b0.x = w0[row +  0]; b0.y = w1[row +  0];
        b1.x = w0[row + 16]; b1.y = w1[row + 16];
        b2.x = w0[row + 32]; b2.y = w1[row + 32];
        b3.x = w0[row + 48]; b3.y = w1[row + 48];
        acc0 = __builtin_amdgcn_wmma_f32_16x16x4_f32(false, a, false, b0, (short)0, acc0, false, false);
        acc1 = __builtin_amdgcn_wmma_f32_16x16x4_f32(false, a, false, b1, (short)0, acc1, false, false);
        acc2 = __builtin_amdgcn_wmma_f32_16x16x4_f32(false, a, false, b2, (short)0, acc2, false, false);
        acc3 = __builtin_amdgcn_wmma_f32_16x16x4_f32(false, a, false, b3, (short)0, acc3, false, false);
    }
    // C/D layout: VGPR r -> M = r + (lane>>4)*8, N = (lane&15) + nt*16
    const int mbase = tile * 16 + (lane >> 4) * 8;
    float* out = Hout + mbase * H + (lane & 15);
    #pragma unroll
    for (int r = 0; r < 8; ++r) {
        out[r * H +  0] = acc0[r];
        out[r * H + 16] = acc1[r];
        out[r * H + 32] = acc2[r];
        out[r * H + 48] = acc3[r];
    }
}

// ---------------- edge scatter: agg[dst] += h[src] * dinv[src]*dinv[dst] ----
// 16 threads per edge, 4 channels each (float4 gather, 4 f32 atomics).
__global__ void k_edge_scatter(const int* __restrict__ ei,
                               const float* __restrict__ dinv,
                               const float* __restrict__ h,
                               float* agg) {
    unsigned tid = blockIdx.x * blockDim.x + threadIdx.x;
    unsigned e = tid >> 4;
    if (e >= NE) return;
    int g = tid & 15;
    int s = ei[e], d = ei[NE + e];
    float nrm = dinv[s] * dinv[d];
    float4 v = *((const float4*)(h + s * H) + g);
    float* ap = agg + d * H + g * 4;
    atomicAdd(ap + 0, v.x * nrm);
    atomicAdd(ap + 1, v.y * nrm);
    atomicAdd(ap + 2, v.z * nrm);
    atomicAdd(ap + 3, v.w * nrm);
}

// ---------------- epilogue: self-loop + bias + BN(eval) + ReLU -------------
__global__ void k_epilogue(const float* __restrict__ agg, const float* __restrict__ h,
                           const float* __restrict__ dinv,
                           const float* __restrict__ b,  const float* __restrict__ g,
                           const float* __restrict__ be, const float* __restrict__ m,
                           const float* __restrict__ v,  float* __restrict__ out) {
    int idx = blockIdx.x * blockDim.x + threadIdx.x;
    if (idx >= NN * H) return;
    int i = idx >> 6, c = idx & 63;
    float di = dinv[i];
    float s = agg[idx] + h[idx] * di * di + b[c];
    float val = (s - m[c]) * (g[c] * rsqrtf(v[c] + EPSV)) + be[c];
    out[idx] = val > 0.0f ? val : 0.0f;
}

// ---------------- pooling ----------------
__global__ void k_count(const int* __restrict__ batch, float* cnt) {
    int i = blockIdx.x * blockDim.x + threadIdx.x;
    if (i < NN) atomicAdd(&cnt[batch[i]], 1.0f);
}
__global__ void k_pool(const float* __restrict__ feat, const int* __restrict__ batch,
                       float* psum, int* pmax) {
    int idx = blockIdx.x * blockDim.x + threadIdx.x;
    if (idx >= NN * H) return;
    int i = idx >> 6, c = idx & 63;
    int bg = batch[i];
    float val = feat[idx];
    atomicAdd(&psum[bg * H + c], val);
    atomicMax(&pmax[bg * H + c], __float_as_int(val));  // val >= 0 post-ReLU
}

// ---------------- final MLP: one 64-thread block per graph -----------------
__global__ __launch_bounds__(64) void k_mlp(const float* __restrict__ psum,
                                            const int* __restrict__ pmax,
                                            const float* __restrict__ cnt,
                                            const float* __restrict__ Wp1,
                                            const float* __restrict__ bp1,
                                            const float* __restrict__ Wp2,
                                            const float* __restrict__ bp2,
                                            float* __restrict__ out) {
    int gr = blockIdx.x, t = threadIdx.x;
    __shared__ float gvec[2 * H];
    __shared__ float hid[H];
    float inv = 1.0f / fmaxf(cnt[gr], 1.0f);
    gvec[t]     = psum[gr * H + t] * inv;
    gvec[H + t] = __int_as_float(pmax[gr * H + t]);
    __syncthreads();
    float acc = bp1[t];
    #pragma unroll 4
    for (int j = 0; j < 2 * H; ++j) acc += gvec[j] * Wp1[j * H + t];
    hid[t] = acc > 0.0f ? acc : 0.0f;
    __syncthreads();
    if (t < 3) {
        float o = bp2[t];
        for (int k = 0; k < H; ++k) o += hid[k] * Wp2[k * 3 + t];
        out[gr * 3 + t] = o;
    }
}

// ---------------- driver ----------------
static void run_layer(const float* feat_in, int in_dim, const float* W, const float* b,
                      const float* g, const float* be, const float* m, const float* v,
                      const int* ei, const float* dinv,
                      float* Hbuf, float* AGG, float* feat_out, hipStream_t stream) {
    if (in_dim == 3)
        k_gemm3<<<(NN * H + 255) / 256, 256, 0, stream>>>(feat_in, W, Hbuf);
    else
        k_gemm64_wmma<<<NN / 16, 32, 0, stream>>>(feat_in, W, Hbuf);
    hipMemsetAsync(AGG, 0, (size_t)NN * H * sizeof(float), stream);
    k_edge_scatter<<<(NE * 16 + 255) / 256, 256, 0, stream>>>(ei, dinv, Hbuf, AGG);
    k_epilogue<<<(NN * H + 255) / 256, 256, 0, stream>>>(AGG, Hbuf, dinv, b, g, be, m, v, feat_out);
}

extern "C" void kernel_launch(void* const* d_in, const int* in_sizes, int n_in,
                              void* d_out, int out_size, void* d_ws, size_t ws_size,
                              hipStream_t stream) {
    const float* x     = (const float*)d_in[0];
    const int*   ei    = (const int*)  d_in[1];
    const int*   batch = (const int*)  d_in[2];
    const float* W1 = (const float*)d_in[3];  const float* b1 = (const float*)d_in[4];
    const float* W2 = (const float*)d_in[5];  const float* b2 = (const float*)d_in[6];
    const float* W3 = (const float*)d_in[7];  const float* b3 = (const float*)d_in[8];
    const float* g1 = (const float*)d_in[9];  const float* be1 = (const float*)d_in[10];
    const float* m1 = (const float*)d_in[11]; const float* v1  = (const float*)d_in[12];
    const float* g2 = (const float*)d_in[13]; const float* be2 = (const float*)d_in[14];
    const float* m2 = (const float*)d_in[15]; const float* v2  = (const float*)d_in[16];
    const float* g3 = (const float*)d_in[17]; const float* be3 = (const float*)d_in[18];
    const float* m3 = (const float*)d_in[19]; const float* v3  = (const float*)d_in[20];
    const float* Wp1 = (const float*)d_in[21]; const float* bp1 = (const float*)d_in[22];
    const float* Wp2 = (const float*)d_in[23]; const float* bp2 = (const float*)d_in[24];
    float* out = (float*)d_out;

    // workspace carve-up (floats): dinv | H | AGG | FEAT | PSUM | PMAX | CNT
    float* ws   = (float*)d_ws;
    float* dinv = ws;
    float* Hbuf = dinv + NN;
    float* AGG  = Hbuf + (size_t)NN * H;
    float* FEAT = AGG  + (size_t)NN * H;
    float* PSUM = FEAT + (size_t)NN * H;
    int*   PMAX = (int*)(PSUM + NG * H);
    float* CNT  = (float*)(PMAX + NG * H);

    // symmetric-norm degrees (graph-constant, computed once per call)
    k_init_deg<<<(NN + 255) / 256, 256, 0, stream>>>(dinv);
    k_deg_accum<<<(NE + 255) / 256, 256, 0, stream>>>(ei, dinv);
    k_dinv<<<(NN + 255) / 256, 256, 0, stream>>>(dinv);

    // three GCN layers (layer1 input is x [N,3])
    run_layer(x,    3, W1, b1, g1, be1, m1, v1, ei, dinv, Hbuf, AGG, FEAT, stream);
    run_layer(FEAT, H, W2, b2, g2, be2, m2, v2, ei, dinv, Hbuf, AGG, FEAT, stream);
    run_layer(FEAT, H, W3, b3, g3, be3, m3, v3, ei, dinv, Hbuf, AGG, FEAT, stream);

    // pooling (sum, max>=0, counts)
    hipMemsetAsync(PSUM, 0, (size_t)(2 * NG * H + NG) * sizeof(float), stream);
    k_count<<<(NN + 255) / 256, 256, 0, stream>>>(batch, CNT);
    k_pool<<<(NN * H + 255) / 256, 256, 0, stream>>>(FEAT, batch, PSUM, PMAX);

    // readout MLP
    k_mlp<<<NG, 64, 0, stream>>>(PSUM, PMAX, CNT, Wp1, bp1, Wp2, bp2, out);
}